// Malleable_Conv2_5D_Depth_63977832842010
// MI455X (gfx1250) — compile-verified
//
#include <hip/hip_runtime.h>
#include <hip/hip_bf16.h>
#include <math.h>

// Problem constants (from reference setup_inputs)
constexpr int N = 4, C = 256, O = 256, H = 64, W = 64;
constexpr int HW = H * W;            // 4096
constexpr int KPROD = 9;
constexpr int KTOT = 3 * KPROD * C;  // 6912 merged-K dimension

typedef __attribute__((ext_vector_type(2))) float v2f;
typedef __attribute__((ext_vector_type(4))) float v4f;
typedef __attribute__((ext_vector_type(8))) float v8f;

// ---------------------------------------------------------------------------
// Prep 1: channels-last transpose  x[n][c][h][w] -> xT[n][h][w][c]
// (so consecutive K values = consecutive channels = contiguous b128 loads)
// ---------------------------------------------------------------------------
__global__ void transpose_x_kernel(const float* __restrict__ x,
                                   float* __restrict__ xT) {
    int idx = blockIdx.x * blockDim.x + threadIdx.x;
    if (idx >= N * C * HW) return;
    int c = idx & (C - 1);
    int rest = idx >> 8;           // / C
    int w = rest & (W - 1);
    rest >>= 6;
    int h = rest & (H - 1);
    int n = rest >> 6;
    // write coalesced (idx consecutive -> c consecutive), read strided
    xT[idx] = x[(((long)n * C + c) * H + h) * W + w];
}

// ---------------------------------------------------------------------------
// Prep 2: pack weights  Apack[o][(b*9+k)*256 + c] = 3*softmax(kw)[b] * W_b[o,c,k]
// ---------------------------------------------------------------------------
__global__ void pack_weights_kernel(const float* __restrict__ w0,
                                    const float* __restrict__ w1,
                                    const float* __restrict__ w2,
                                    const float* __restrict__ kw,
                                    float* __restrict__ Apack) {
    long idx = (long)blockIdx.x * blockDim.x + threadIdx.x;
    if (idx >= (long)O * KTOT) return;
    int kk = (int)(idx % KTOT);
    int o  = (int)(idx / KTOT);
    int b  = kk / (KPROD * C);          // 0..2
    int k  = (kk >> 8) % KPROD;         // 0..8
    int c  = kk & (C - 1);
    float m = fmaxf(kw[0], fmaxf(kw[1], kw[2]));
    float e0 = expf(kw[0] - m), e1 = expf(kw[1] - m), e2 = expf(kw[2] - m);
    float s = e0 + e1 + e2;
    float scale = 3.0f * ((b == 0 ? e0 : (b == 1 ? e1 : e2)) / s);
    const float* wp = (b == 0) ? w0 : (b == 1 ? w1 : w2);
    Apack[idx] = scale * wp[((long)(o * C + c)) * KPROD + k];
}

// ---------------------------------------------------------------------------
// Prep 3: depth-class masks  maskbuf[((n*3+b)*9+k)*4096 + m]
//   = invalid ? 0.2 : softmax_class[b+1]   (faithful to reference)
// ---------------------------------------------------------------------------
__global__ void mask_kernel(const float* __restrict__ depth,
                            const float* __restrict__ fx,
                            const float* __restrict__ anchor,   // 5 floats
                            const float* __restrict__ temperature,
                            float* __restrict__ maskbuf) {
    int idx = blockIdx.x * blockDim.x + threadIdx.x;
    if (idx >= N * HW) return;
    int m = idx & (HW - 1);
    int n = idx >> 12;
    int h = m >> 6, w = m & (W - 1);

    float d[KPROD];
#pragma unroll
    for (int ki = 0; ki < 3; ++ki)
#pragma unroll
        for (int kj = 0; kj < 3; ++kj) {
            int hh = h + ki - 1, ww = w + kj - 1;
            bool inb = ((unsigned)hh < (unsigned)H) && ((unsigned)ww < (unsigned)W);
            d[ki * 3 + kj] = inb ? depth[(long)n * HW + hh * W + ww] : 0.0f;
        }

    float dc = d[4];
    float validc = (dc != 0.0f) ? 1.0f : 0.0f;
    float center_depth = dc * validc;
    float grid_range = center_depth / fx[n];                  // PIXEL*DILATION = 1
    float inv_gr = 100.0f / (1e-5f + grid_range * 100.0f);
    float denom = 1e-5f + fmaxf(temperature[0], 0.0f);
    float a0 = anchor[0], a4 = anchor[4];

#pragma unroll
    for (int k = 0; k < KPROD; ++k) {
        float val  = (d[k] != 0.0f) ? validc : 0.0f;
        float dcol = d[k] * val;
        float diff = dcol - center_depth;
        float rel  = diff * inv_gr;
        float lg[5];
#pragma unroll
        for (int a = 0; a < 5; ++a) {
            float t = rel - anchor[a];
            lg[a] = -(t * t) / denom;
        }
        if (diff < a0) lg[0] = -lg[0];   // *(1 - 2*out0)
        if (diff > a4) lg[4] = -lg[4];   // *(1 - 2*out4)
        float mx = lg[0];
#pragma unroll
        for (int a = 1; a < 5; ++a) mx = fmaxf(mx, lg[a]);
        float e[5], s = 0.0f;
#pragma unroll
        for (int a = 0; a < 5; ++a) { e[a] = expf(lg[a] - mx); s += e[a]; }
        float inv_s = 1.0f / s;
#pragma unroll
        for (int b = 0; b < 3; ++b) {
            float mv = (val == 0.0f) ? 0.2f : e[b + 1] * inv_s;
            maskbuf[(((long)n * 3 + b) * KPROD + k) * HW + m] = mv;
        }
    }
}

// ---------------------------------------------------------------------------
// Main GEMM: out[n][o][m] = sum_kk Apack[o][kk] * (xT * mask)[kk][m]
// Block = 256 threads = 8 wave32s.  Wave -> (o_quad 0..3, m_sub 0..1):
//   covers o = o_quad*64 .. +63 (4 WMMA o-tiles) x 16 pixels.
// Block covers full O=256 x 32 pixels for one n.  Grid = 4 * 128 = 512.
//
// K mapping trick: WMMA sums all 4 K-slots, so any per-step assignment of
// channels to (lane_hi, vgpr) slots is valid as long as A and B agree.
// Per 8-channel group we use K-sets {c,c+1,c+4,c+5} then {c+2,c+3,c+6,c+7},
// so each lane's 4 channels (c + lane_hi*4 .. +3) come from ONE b128 load
// that feeds TWO wmma ops  ->  5 vmem per 8 wmma.
// ---------------------------------------------------------------------------
__global__ __launch_bounds__(256) void
gemm_wmma_kernel(const float* __restrict__ xT,
                 const float* __restrict__ Apack,
                 const float* __restrict__ maskbuf,
                 float* __restrict__ out) {
    const int lane    = threadIdx.x & 31;
    const int wv      = threadIdx.x >> 5;       // 0..7
    const int lane_lo = lane & 15;              // N column / A row within tile
    const int lane_hi = lane >> 4;              // K-slot selector (0/1)
    const int n       = blockIdx.x >> 7;        // / 128
    const int mpair   = blockIdx.x & 127;
    const int o_quad  = wv & 3;
    const int m_sub   = wv >> 2;
    const int m0      = (mpair * 2 + m_sub) * 16;
    const int o_base  = o_quad * 64;
    const int m_lane  = m0 + lane_lo;
    const int h = m_lane >> 6, w = m_lane & (W - 1);

    v8f acc[4] = {v8f{}, v8f{}, v8f{}, v8f{}};

    const float* xTn = xT + (long)n * HW * C;

    for (int b = 0; b < 3; ++b) {
#pragma unroll
        for (int k = 0; k < KPROD; ++k) {
            const int ki = k / 3, kj = k - ki * 3;
            const int h2 = h + ki - 1, w2 = w + kj - 1;
            const bool inb = ((unsigned)h2 < (unsigned)H) &&
                             ((unsigned)w2 < (unsigned)W);
            // zero-padded x => contribution 0; drop via mv = 0, clamp address
            const float mv = inb
                ? maskbuf[(((long)n * 3 + b) * KPROD + k) * HW + m_lane]
                : 0.0f;
            const int h2c = inb ? h2 : 0, w2c = inb ? w2 : 0;
            const float* xp = xTn + ((h2c << 6) + w2c) * C + (lane_hi << 2);
            const float* ap = Apack + (long)(o_base + lane_lo) * KTOT
                              + ((b * KPROD + k) << 8) + (lane_hi << 2);
#pragma unroll 2
            for (int c = 0; c < C; c += 8) {
                v4f bx = *(const v4f*)(xp + c);          // 4 channels, 1 b128
                v4f a0 = *(const v4f*)(ap + c);
                v4f a1 = *(const v4f*)(ap + 16 * KTOT + c);
                v4f a2 = *(const v4f*)(ap + 32 * KTOT + c);
                v4f a3 = *(const v4f*)(ap + 48 * KTOT + c);
                v4f bf = bx * mv;                        // 2x v_pk_mul_f32
                v2f bf_lo = __builtin_shufflevector(bf, bf, 0, 1);
                v2f bf_hi = __builtin_shufflevector(bf, bf, 2, 3);
                v2f a0lo = __builtin_shufflevector(a0, a0, 0, 1);
                v2f a1lo = __builtin_shufflevector(a1, a1, 0, 1);
                v2f a2lo = __builtin_shufflevector(a2, a2, 0, 1);
                v2f a3lo = __builtin_shufflevector(a3, a3, 0, 1);
                acc[0] = __builtin_amdgcn_wmma_f32_16x16x4_f32(
                    false, a0lo, false, bf_lo, (short)0, acc[0], false, false);
                acc[1] = __builtin_amdgcn_wmma_f32_16x16x4_f32(
                    false, a1lo, false, bf_lo, (short)0, acc[1], false, false);
                acc[2] = __builtin_amdgcn_wmma_f32_16x16x4_f32(
                    false, a2lo, false, bf_lo, (short)0, acc[2], false, false);
                acc[3] = __builtin_amdgcn_wmma_f32_16x16x4_f32(
                    false, a3lo, false, bf_lo, (short)0, acc[3], false, false);
                v2f a0hi = __builtin_shufflevector(a0, a0, 2, 3);
                v2f a1hi = __builtin_shufflevector(a1, a1, 2, 3);
                v2f a2hi = __builtin_shufflevector(a2, a2, 2, 3);
                v2f a3hi = __builtin_shufflevector(a3, a3, 2, 3);
                acc[0] = __builtin_amdgcn_wmma_f32_16x16x4_f32(
                    false, a0hi, false, bf_hi, (short)0, acc[0], false, false);
                acc[1] = __builtin_amdgcn_wmma_f32_16x16x4_f32(
                    false, a1hi, false, bf_hi, (short)0, acc[1], false, false);
                acc[2] = __builtin_amdgcn_wmma_f32_16x16x4_f32(
                    false, a2hi, false, bf_hi, (short)0, acc[2], false, false);
                acc[3] = __builtin_amdgcn_wmma_f32_16x16x4_f32(
                    false, a3hi, false, bf_hi, (short)0, acc[3], false, false);
            }
        }
    }

    // C/D layout: VGPR r, lanes 0-15 -> M=r, lanes 16-31 -> M=8+r; N = lane%16
    float* op = out + (long)n * O * HW + m0 + lane_lo;
#pragma unroll
    for (int t = 0; t < 4; ++t) {
        const int ob = o_base + t * 16 + lane_hi * 8;
#pragma unroll
        for (int r = 0; r < 8; ++r)
            op[(long)(ob + r) * HW] = acc[t][r];
    }
}

// ---------------------------------------------------------------------------
// Launch
// ---------------------------------------------------------------------------
extern "C" void kernel_launch(void* const* d_in, const int* in_sizes, int n_in,
                              void* d_out, int out_size, void* d_ws, size_t ws_size,
                              hipStream_t stream) {
    const float* x      = (const float*)d_in[0];
    const float* depth  = (const float*)d_in[1];
    const float* fx     = (const float*)d_in[2];
    const float* w0     = (const float*)d_in[3];
    const float* w1     = (const float*)d_in[4];
    const float* w2     = (const float*)d_in[5];
    const float* anchor = (const float*)d_in[6];
    const float* temp   = (const float*)d_in[7];
    const float* kw     = (const float*)d_in[8];
    float* out = (float*)d_out;

    // workspace layout (floats): xT | Apack | maskbuf  (~25.6 MB total)
    float* ws      = (float*)d_ws;
    float* xT      = ws;                                   // 4*64*64*256
    float* Apack   = ws + (long)N * HW * C;                // 256*6912
    float* maskbuf = Apack + (long)O * KTOT;               // 4*3*9*4096

    {   // channels-last transpose of x
        int total = N * C * HW;
        transpose_x_kernel<<<(total + 255) / 256, 256, 0, stream>>>(x, xT);
    }
    {   // weight packing (3 branches merged, kernel_weight softmax folded in)
        long total = (long)O * KTOT;
        pack_weights_kernel<<<(int)((total + 255) / 256), 256, 0, stream>>>(
            w0, w1, w2, kw, Apack);
    }
    {   // depth-class masks
        int total = N * HW;
        mask_kernel<<<(total + 255) / 256, 256, 0, stream>>>(
            depth, fx, anchor, temp, maskbuf);
    }
    // main WMMA GEMM: grid = N * (HW/16 tiles / 2 per block) = 4 * 128
    gemm_wmma_kernel<<<N * 128, 256, 0, stream>>>(xT, Apack, maskbuf, out);
}